// EquivariantConvExp_25237227831586
// MI455X (gfx1250) — compile-verified
//
#include <hip/hip_runtime.h>

typedef __attribute__((ext_vector_type(2))) float v2f;
typedef __attribute__((ext_vector_type(4))) float v4f;
typedef __attribute__((ext_vector_type(8))) float v8f;

#define N_BLOCKS 8
#define N_BASIS  3
#define N_TERMS  10

// ---------------------------------------------------------------------------
// Setup kernel (1 thread): fold the 8 per-block truncated matrix exponentials
// into one 2x2 matrix T and the scalar log-det L.  ~1 KFLOP total.
// ws[0..3] = T row-major, ws[4] = L.
// ---------------------------------------------------------------------------
__global__ void eqconv_setup_kernel(const float* __restrict__ w,
                                    const float* __restrict__ basis,
                                    float* __restrict__ ws) {
    if (threadIdx.x != 0 || blockIdx.x != 0) return;

    float T00 = 1.f, T01 = 0.f, T10 = 0.f, T11 = 1.f;   // running product
    float Lsum = 0.f;

    // reference applies blocks i = 7 .. 0, z <- E_i z, so T = E0*E1*...*E7,
    // built by left-multiplying while walking i = 7 .. 0.
    for (int i = N_BLOCKS - 1; i >= 0; --i) {
        // filt middle row from the steerable basis expansion:
        // flat 3x3 indices: [1,0]=3, [1,1]=4, [1,2]=5
        float fl = 0.f, fc = 0.f, fr = 0.f;
        for (int nb = 0; nb < N_BASIS; ++nb) {
            const float wn = w[i * N_BASIS + nb];
            const float* bp = basis + (size_t)(i * N_BASIS + nb) * 9;
            fl += wn * bp[3];
            fc += wn * bp[4];
            fr += wn * bp[5];
        }
        Lsum += fc;

        // M = [[c, r], [l, c]];  E = sum_{k=0..10} M^k / k!
        float t00 = 1.f, t01 = 0.f, t10 = 0.f, t11 = 1.f;  // term
        float e00 = 1.f, e01 = 0.f, e10 = 0.f, e11 = 1.f;  // accumulator
        for (int k = 1; k <= N_TERMS; ++k) {
            const float inv_k = 1.f / (float)k;
            const float n00 = (fc * t00 + fr * t10) * inv_k;
            const float n01 = (fc * t01 + fr * t11) * inv_k;
            const float n10 = (fl * t00 + fc * t10) * inv_k;
            const float n11 = (fl * t01 + fc * t11) * inv_k;
            t00 = n00; t01 = n01; t10 = n10; t11 = n11;
            e00 += t00; e01 += t01; e10 += t10; e11 += t11;
        }

        // T = E * T
        const float m00 = e00 * T00 + e01 * T10;
        const float m01 = e00 * T01 + e01 * T11;
        const float m10 = e10 * T00 + e11 * T10;
        const float m11 = e10 * T01 + e11 * T11;
        T00 = m00; T01 = m01; T10 = m10; T11 = m11;
    }

    ws[0] = T00; ws[1] = T01; ws[2] = T10; ws[3] = T11;
    ws[4] = -2.f * Lsum;   // hw = 2.0, per-block trace = fc
}

// ---------------------------------------------------------------------------
// Streaming kernel.  Per wave iteration: 4 chunks x 32 samples through four
// V_WMMA_F32_16X16X4_F32 ops.  A(16x4) carries T block-diagonally (rows 0..3),
// B(4x16) carries 2 samples per column, D rows 0..3 give both samples'
// outputs per column.  4 independent b64 loads in flight per iteration;
// all traffic non-temporal (zero reuse).  Memory-bound: ~80 MB total
// -> ~3.4 us at 23.3 TB/s.
// ---------------------------------------------------------------------------
__global__ void eqconv_apply_kernel(const float* __restrict__ x,
                                    const float* __restrict__ tw,
                                    float* __restrict__ out_z,
                                    float* __restrict__ out_ld,
                                    unsigned n) {
    const float T00 = tw[0], T01 = tw[1], T10 = tw[2], T11 = tw[3];
    const float L = tw[4];

    const unsigned lane   = threadIdx.x & 31u;
    const unsigned col    = lane & 15u;     // column in D / sample-pair index
    const unsigned half   = lane >> 4;      // 0: rows K0/K1, 1: rows K2/K3
    const unsigned waveId = (blockIdx.x * blockDim.x + threadIdx.x) >> 5;
    const unsigned nWaves = (gridDim.x * blockDim.x) >> 5;

    // A matrix (16x4 f32): VGPR0 = K0|K2, VGPR1 = K1|K3 (lanes 0-15|16-31)
    // A[0,0:2)=T row0, A[1,0:2)=T row1, A[2,2:4)=T row0, A[3,2:4)=T row1
    v2f a; a.x = 0.f; a.y = 0.f;
    if (lane == 0u || lane == 18u) { a.x = T00; a.y = T01; }
    if (lane == 1u || lane == 19u) { a.x = T10; a.y = T11; }

    const unsigned chunks = n >> 5;          // 32 samples per chunk
    const unsigned groups = chunks >> 2;     // 128 samples per group

    for (unsigned g = waveId; g < groups; g += nWaves) {
        const unsigned S0   = g << 7;                  // first sample of group
        const unsigned base = S0 + 2u * col + half;    // this lane's sample, u=0

        // 4 independent loads issued before any use -> 4 loads in flight
        v2f bm[4];
#pragma unroll
        for (unsigned u = 0; u < 4u; ++u)
            bm[u] = __builtin_nontemporal_load(
                        (const v2f*)(x + 2u * (size_t)(base + (u << 5))));

        v8f d[4];
#pragma unroll
        for (unsigned u = 0; u < 4u; ++u) {
            v8f c = {};
            d[u] = __builtin_amdgcn_wmma_f32_16x16x4_f32(
                       /*neg_a=*/false, a, /*neg_b=*/false, bm[u],
                       /*c_mod=*/(short)0, c, /*reuse_a=*/false, /*reuse_b=*/false);
        }

        if (half == 0u) {
            // D rows 0..3 (VGPR0..3, lanes 0-15): two consecutive samples
#pragma unroll
            for (unsigned u = 0; u < 4u; ++u) {
                v4f o = { d[u][0], d[u][1], d[u][2], d[u][3] };
                __builtin_nontemporal_store(
                    o, (v4f*)(out_z + (size_t)2u * (S0 + (u << 5))
                                     + 4u * (size_t)col));
            }
        }

        // broadcast scalar: one b128 per lane covers all 128 samples of group
        v4f ld4 = { L, L, L, L };
        __builtin_nontemporal_store(ld4, (v4f*)(out_ld + (size_t)S0 + 4u * lane));
    }

    // scalar tail for samples not covered by full groups
    // (empty for B = 4194304 = 32768 * 128)
    const unsigned done   = groups << 7;
    const unsigned tid    = blockIdx.x * blockDim.x + threadIdx.x;
    const unsigned stride = gridDim.x * blockDim.x;
    for (unsigned i = done + tid; i < n; i += stride) {
        const float z0 = x[2u * (size_t)i];
        const float z1 = x[2u * (size_t)i + 1u];
        out_z[2u * (size_t)i]      = T00 * z0 + T01 * z1;
        out_z[2u * (size_t)i + 1u] = T10 * z0 + T11 * z1;
        out_ld[i] = L;
    }
}

extern "C" void kernel_launch(void* const* d_in, const int* in_sizes, int n_in,
                              void* d_out, int out_size, void* d_ws, size_t ws_size,
                              hipStream_t stream) {
    const float* x     = (const float*)d_in[0];   // [B,2]
    const float* w     = (const float*)d_in[1];   // [8,3]
    const float* basis = (const float*)d_in[2];   // [8,3,1,1,3,3]
    const unsigned n = (unsigned)(in_sizes[0] / 2);   // B samples

    float* ws     = (float*)d_ws;                 // T (4 floats) + L (1 float)
    float* out_z  = (float*)d_out;                // [B,2]
    float* out_ld = out_z + (size_t)2 * n;        // [B]

    eqconv_setup_kernel<<<1, 1, 0, stream>>>(w, basis, ws);

    // 2048 blocks * 8 waves = 16384 waves; 32768 groups -> 2 groups/wave
    eqconv_apply_kernel<<<2048, 256, 0, stream>>>(x, ws, out_z, out_ld, n);
}